// S4DDendrite_18769007084375
// MI455X (gfx1250) — compile-verified
//
#include <hip/hip_runtime.h>
#include <hip/hip_bf16.h>

// ---------------------------------------------------------------------------
// S4D bidirectional conv as per-channel Toeplitz GEMM on the CDNA5 WMMA pipe.
//   B=8, L=2048, D=512, N=32, DIRS=2
//   Y_d = M_d X_d,  M_d[t,s] = ke_d[t - s + 2047]   (fp32 WMMA 16x16x4)
// X_d half-tiles (strided 4B gather) are staged into LDS by the Tensor Data
// Mover (TENSOR_LOAD_TO_LDS + s_wait_tensorcnt) instead of vector loads.
// ---------------------------------------------------------------------------

#define S4D_B 8
#define S4D_L 2048
#define S4D_D 512
#define S4D_N 32

typedef __attribute__((ext_vector_type(2))) float v2f;
typedef __attribute__((ext_vector_type(8))) float v8f;
typedef __attribute__((ext_vector_type(4))) unsigned int u32x4;
typedef __attribute__((ext_vector_type(4))) int i32x4;
typedef __attribute__((ext_vector_type(8))) int i32x8;

// ---------------------------------------------------------------------------
// Kernel 1: build extended kernels ke[d][4096]
//   ke[d][j] = (j>=2047 ? k_fwd[j-2047] : 0) + (j<=2047 ? k_bwd[2047-j] : 0)
//   k_dir[t] = 2 * Re( sum_n scaled[dir,d,n] * exp(delta_a[d,n] * t) )
// ---------------------------------------------------------------------------
__global__ __launch_bounds__(256) void s4d_build_kernels(
    const float* __restrict__ log_dt,      // (D)
    const float* __restrict__ log_a_real,  // (D,N)
    const float* __restrict__ a_imag,      // (D,N)
    const float* __restrict__ coeffs,      // (2,D,N,2) view_as_real
    float* __restrict__ ke)                // (D,4096) workspace
{
  __shared__ float s_dar[2 * S4D_N];
  __shared__ float s_dai[2 * S4D_N];
  __shared__ float s_sr[2 * S4D_N];
  __shared__ float s_si[2 * S4D_N];

  const int d   = blockIdx.x;
  const int tid = threadIdx.x;

  if (tid < 2 * S4D_N) {
    const int dir = tid >> 5;
    const int n   = tid & (S4D_N - 1);
    const float dt = expf(log_dt[d]);
    const float ar = -expf(log_a_real[d * S4D_N + n]);
    const float ai = a_imag[d * S4D_N + n];
    const float dar = ar * dt;
    const float dai = ai * dt;
    // e = exp(dar + i*dai)
    const float em = expf(dar);
    const float ec = em * cosf(dai) - 1.0f;   // Re(e) - 1
    const float es = em * sinf(dai);          // Im(e)
    // g = (e-1)/a = (e-1) * conj(a) / |a|^2
    const float den = ar * ar + ai * ai;
    const float gr = (ec * ar + es * ai) / den;
    const float gi = (es * ar - ec * ai) / den;
    // scaled = c * g
    const float cr = coeffs[((dir * S4D_D + d) * S4D_N + n) * 2 + 0];
    const float ci = coeffs[((dir * S4D_D + d) * S4D_N + n) * 2 + 1];
    s_dar[tid] = dar;
    s_dai[tid] = dai;
    s_sr[tid]  = cr * gr - ci * gi;
    s_si[tid]  = cr * gi + ci * gr;
  }
  __syncthreads();

  for (int j = tid; j < 2 * S4D_L; j += 256) {
    const int off = j - (S4D_L - 1);
    float val = 0.0f;
    if (off >= 0 && off <= S4D_L - 1) {          // forward direction (dir 0)
      const float tt = (float)off;
      #pragma unroll 4
      for (int n = 0; n < S4D_N; ++n) {
        val += 2.0f * expf(s_dar[n] * tt) *
               (s_sr[n] * cosf(s_dai[n] * tt) - s_si[n] * sinf(s_dai[n] * tt));
      }
    }
    if (off <= 0) {                               // backward direction (dir 1)
      const float tt = (float)(-off);
      #pragma unroll 4
      for (int n = S4D_N; n < 2 * S4D_N; ++n) {
        val += 2.0f * expf(s_dar[n] * tt) *
               (s_sr[n] * cosf(s_dai[n] * tt) - s_si[n] * sinf(s_dai[n] * tt));
      }
    }
    ke[(size_t)d * (2 * S4D_L) + j] = val;
  }
}

// ---------------------------------------------------------------------------
// Kernel 2: per-channel Toeplitz GEMM with V_WMMA_F32_16X16X4_F32.
// grid = (L/64, D); block = 128 threads = 4 wave32s; each wave owns a 16-row
// tile. LDS: full ke_d (16KB, b128 vector loads) + half of X_d (32KB, staged
// by the TDM as a (1 x 8 x 1024) strided tile). Two accumulator chains (K
// split) hide the WMMA->WMMA RAW latency.
// ---------------------------------------------------------------------------
__global__ __launch_bounds__(128) void s4d_conv_wmma(
    const float* __restrict__ in,          // (B,L,D)
    const float* __restrict__ ke,          // (D,4096)
    const float* __restrict__ input_scale, // (D)
    float* __restrict__ out)               // (B,L,D)
{
  __shared__ __align__(16) float ke_s[2 * S4D_L];         // 16 KB
  __shared__ __align__(16) float xs[(S4D_L / 2) * S4D_B]; // 32 KB: [s_local][b]

  const int tid  = threadIdx.x;
  const int lane = tid & 31;
  const int wave = tid >> 5;
  const int d    = blockIdx.y;
  const int r0   = blockIdx.x * 64 + wave * 16;   // row-tile base (time t)

  // stage the extended kernel for this channel (contiguous -> b128 loads)
  {
    const float4* src = reinterpret_cast<const float4*>(ke + (size_t)d * (2 * S4D_L));
    float4* dst = reinterpret_cast<float4*>(ke_s);
    for (int idx = tid; idx < (2 * S4D_L) / 4; idx += 128)
      dst[idx] = src[idx];
  }

  // per-lane fragment coordinates (ISA 7.12.2, 32-bit A 16x4 / B 4x16 layout)
  const int m   = lane & 15;         // A row within tile  -> t = r0 + m
  const int col = lane & 15;         // B column (batch)
  const int kk  = (lane >> 4) << 1;  // K sub-offset: lanes 16-31 hold K=2,3
  const int t   = r0 + m;

  v8f acc0 = {};
  v8f acc1 = {};

  for (int h = 0; h < 2; ++h) {
    __syncthreads();                 // previous half's xs reads complete
    const int sbase = h << 10;       // global K base of this stage

    // ---- TDM: DMA-gather X_d[sbase..sbase+1024) x B into LDS -------------
    // Tensor descriptor (ISA 8.3-8.6):
    //   data_size = 4B; tile = (x=1, y=B=8, z=1024)
    //   tensor_dim0_stride = L*D (between batches), tensor_dim1_stride = D
    //   LDS fill order x->y->z  ==>  xs[s*8 + b]
    if (wave == 0) {
      const uint64_t ga =
          (uint64_t)(uintptr_t)(in + ((size_t)sbase * S4D_D + d));
      const unsigned int lds_off = (unsigned int)(uintptr_t)(&xs[0]);

      u32x4 g0;
      g0[0] = 1u;                                  // count=1, user descriptor
      g0[1] = lds_off;                             // lds_addr
      g0[2] = (unsigned int)(ga & 0xffffffffu);    // global_addr[31:0]
      g0[3] = (unsigned int)((ga >> 32) & 0x01ffffffu) | (2u << 30); // [56:32], type=2

      const unsigned int td0 = 1u;                 // tensor_dim0 (x length)
      const unsigned int td1 = (unsigned int)S4D_B;        // tensor_dim1
      const unsigned long long s0 = (unsigned long long)S4D_L * S4D_D; // dim0 stride
      const unsigned long long s1 = (unsigned long long)S4D_D;         // dim1 stride
      i32x8 g1;
      g1[0] = (int)(2u << 16);                     // wg_mask=0, data_size=2 (4B)
      g1[1] = (int)((td0 & 0xffffu) << 16);        // atomic_addr=0, td0[15:0]
      g1[2] = (int)(((td0 >> 16) & 0xffffu) | ((td1 & 0xffffu) << 16));
      g1[3] = (int)(((td1 >> 16) & 0xffffu) | (1u << 16));   // tile_dim0 = 1
      g1[4] = (int)((unsigned)S4D_B | (1024u << 16));        // tile_dim1, tile_dim2
      g1[5] = (int)(unsigned int)(s0 & 0xffffffffu);         // td0_stride[31:0]
      g1[6] = (int)((unsigned int)((s0 >> 32) & 0xffffu) |
                    (unsigned int)((s1 & 0xffffu) << 16));   // td0_str[47:32], td1_str[15:0]
      g1[7] = (int)(unsigned int)((s1 >> 16) & 0xffffffffu); // td1_stride[47:16]

      i32x4 g2;
      g2[0] = 1024;                                // tensor_dim2 (z length)
      g2[1] = 0;                                   // tensor_dim3
      g2[2] = 0;                                   // tensor_dim2_stride lo
      g2[3] = 0;                                   // stride hi / tile_dim3=0
      i32x4 g3 = {0, 0, 0, 0};
      i32x8 gx = {0, 0, 0, 0, 0, 0, 0, 0};         // extension arg (clang-23): unused

      __builtin_amdgcn_tensor_load_to_lds(g0, g1, g2, g3, gx, 0);
      __builtin_amdgcn_s_wait_tensorcnt(0);
    }
    __syncthreads();                 // xs ready for all 4 waves

    #pragma unroll 4
    for (int i = 0; i < 512; i += 4) {
      { // chain 0: K in [sbase, sbase+512)
        const int sl = i + kk;                       // local s of this lane's K0
        const float b0 = (col < S4D_B) ? xs[(sl    ) * S4D_B + col] : 0.0f;
        const float b1 = (col < S4D_B) ? xs[(sl + 1) * S4D_B + col] : 0.0f;
        const int ia = t - (sbase + sl) + (S4D_L - 1);
        v2f av = { ke_s[ia], ke_s[ia - 1] };         // M[t, s], M[t, s+1]
        v2f bv = { b0, b1 };
        acc0 = __builtin_amdgcn_wmma_f32_16x16x4_f32(
            false, av, false, bv, (short)0, acc0, false, false);
      }
      { // chain 1: K in [sbase+512, sbase+1024)
        const int sl = 512 + i + kk;
        const float b0 = (col < S4D_B) ? xs[(sl    ) * S4D_B + col] : 0.0f;
        const float b1 = (col < S4D_B) ? xs[(sl + 1) * S4D_B + col] : 0.0f;
        const int ia = t - (sbase + sl) + (S4D_L - 1);
        v2f av = { ke_s[ia], ke_s[ia - 1] };
        v2f bv = { b0, b1 };
        acc1 = __builtin_amdgcn_wmma_f32_16x16x4_f32(
            false, av, false, bv, (short)0, acc1, false, false);
      }
    }
  }

  // epilogue: C/D layout -> VGPR r holds M = r + 8*(lane/16), N = lane%16
  const float scale = input_scale[d];
  const int b = lane & 15;
  if (b < S4D_B) {
    const int thi = (lane >> 4) * 8;
    #pragma unroll
    for (int r = 0; r < 8; ++r) {
      const int tt = r0 + r + thi;
      const size_t o = ((size_t)b * S4D_L + tt) * S4D_D + d;
      out[o] = acc0[r] + acc1[r] + in[o] * scale;
    }
  }
}

// ---------------------------------------------------------------------------
// Host launcher
// ---------------------------------------------------------------------------
extern "C" void kernel_launch(void* const* d_in, const int* in_sizes, int n_in,
                              void* d_out, int out_size, void* d_ws, size_t ws_size,
                              hipStream_t stream) {
  const float* inputs      = (const float*)d_in[0];  // (B,L,D)
  const float* log_dt      = (const float*)d_in[1];  // (D)
  const float* log_a_real  = (const float*)d_in[2];  // (D,N)
  const float* a_imag      = (const float*)d_in[3];  // (D,N)
  const float* coeffs      = (const float*)d_in[4];  // (2,D,N,2)
  const float* input_scale = (const float*)d_in[5];  // (D)
  float* out = (float*)d_out;
  float* ke  = (float*)d_ws;                         // D * 4096 floats = 8 MB

  s4d_build_kernels<<<dim3(S4D_D), dim3(256), 0, stream>>>(
      log_dt, log_a_real, a_imag, coeffs, ke);

  s4d_conv_wmma<<<dim3(S4D_L / 64, S4D_D), dim3(128), 0, stream>>>(
      inputs, ke, input_scale, out);
}